// GMKT_67267777790123
// MI455X (gfx1250) — compile-verified
//
#include <hip/hip_runtime.h>
#include <math.h>

typedef float v2f __attribute__((ext_vector_type(2)));
typedef float v8f __attribute__((ext_vector_type(8)));

// Explicit global-address-space pointers: force global_load_* / global_store_*
// (flat_load ticks both LOADcnt and DScnt on CDNA5 and serializes against the
// LDS traffic in the latency-critical scan loop).
typedef const __attribute__((address_space(1))) float* gcfloat;
typedef const __attribute__((address_space(1))) int*   gcint;
typedef __attribute__((address_space(1))) float*       gfloat;

#define Bsz  64
#define Ssz  200
#define Ksz  32
#define Esz  64
#define Csz  50
#define Tlen 198          // S - 2
#define VSTR 68           // LDS row stride for V (pad to avoid bank conflicts)

// -------------------------------------------------------------------------
// Kernel 1: per-token precompute.
// One 64-thread block per (b,s) token; thread t owns feature dim t.
// Produces (all strided by 64 per token, attention rows zero-padded 50..63):
//   Wo   = softmax attention w          (used at s = t+1)
//   WNo  = softmax attention w_next     (used at s = t+2, from raw q_emb)
//   WTYo = softmax attention w_type     (used at s = t+1)
//   Eo   = erase gate sigmoid, ADo = add gate tanh, QEo = q embedding copy
// -------------------------------------------------------------------------
__global__ __launch_bounds__(64) void gmkt_pre(
    const int* __restrict__ qd,  const int* __restrict__ adat,
    const int* __restrict__ ldat, const int* __restrict__ ddat,
    const int* __restrict__ qqn, const int* __restrict__ qln,
    const int* __restrict__ lqn, const int* __restrict__ lln,
    const float* __restrict__ q_embed, const float* __restrict__ l_embed,
    const float* __restrict__ a_embed, const float* __restrict__ key_matrix,
    const float* __restrict__ GQQ, const float* __restrict__ GQL,
    const float* __restrict__ GLL, const float* __restrict__ GLQ,
    const float* __restrict__ GQw, const float* __restrict__ GQb,
    const float* __restrict__ GLw, const float* __restrict__ GLb,
    const float* __restrict__ mQkw, const float* __restrict__ mQkb,
    const float* __restrict__ mLkw, const float* __restrict__ mLkb,
    const float* __restrict__ mQtw, const float* __restrict__ mQtb,
    const float* __restrict__ mLtw, const float* __restrict__ mLtb,
    const float* __restrict__ eQw, const float* __restrict__ eQb,
    const float* __restrict__ eLw, const float* __restrict__ eLb,
    const float* __restrict__ aQw, const float* __restrict__ aQb,
    const float* __restrict__ aLw, const float* __restrict__ aLb,
    float* __restrict__ Wo, float* __restrict__ WNo, float* __restrict__ WTYo,
    float* __restrict__ Eo, float* __restrict__ ADo, float* __restrict__ QEo)
{
    const int tok = blockIdx.x;
    const int t   = threadIdx.x;

    __shared__ int   nb[Ksz];
    __shared__ float sx[128];     // shared mat-vec input (up to 128 wide)
    __shared__ float kvS[64];
    __shared__ float lg[64];
    __shared__ float red[2];

    const int iq = ((gcint)qd)[tok];
    const int il = ((gcint)ldat)[tok];
    const int ia = ((gcint)adat)[tok];
    const int id = ((gcint)ddat)[tok];   // material type, uniform per block

    const float qe = ((gcfloat)q_embed)[iq * Esz + t];
    const float ae = ((gcfloat)a_embed)[ia * Esz + t];
    const float le = ((gcfloat)l_embed)[il * Esz + t];

    // neighbor mean + linear: returns component t of (mean @ W.T)
    auto agg = [&](const int* tbl, const float* emb_, int idx,
                   const float* W_) -> float {
        gcfloat emb = (gcfloat)emb_;
        gcfloat W   = (gcfloat)W_;
        gcint   tb  = (gcint)tbl;
        __syncthreads();
        if (t < Ksz) nb[t] = tb[idx * Ksz + t];
        __syncthreads();
        int cnt = 0; float s = 0.f;
        #pragma unroll 8
        for (int k = 0; k < Ksz; ++k) {
            const int n = nb[k];
            cnt += (n != 0);
            s += emb[n * Esz + t];
        }
        const float mean = (cnt > 0) ? s / (float)cnt : 0.f;
        __syncthreads();
        sx[t] = mean;
        __syncthreads();
        float y = 0.f;
        #pragma unroll 8
        for (int j = 0; j < 64; ++j) y += W[t * 64 + j] * sx[j];
        return y;
    };

    const float qq = agg(qqn, q_embed, iq, GQQ);
    const float ql = agg(qln, l_embed, iq, GQL);
    const float lq = agg(lqn, q_embed, il, GLQ);
    const float ll = agg(lln, l_embed, il, GLL);

    // qt = GNN_Q(q_emb + 0.5 qq + 0.5 ql), lt = GNN_L(l_emb + 0.5 ll + 0.5 lq)
    __syncthreads(); sx[t] = qe + 0.5f * qq + 0.5f * ql; __syncthreads();
    float qt = ((gcfloat)GQb)[t];
    #pragma unroll 8
    for (int j = 0; j < 64; ++j) qt += ((gcfloat)GQw)[t * 64 + j] * sx[j];
    __syncthreads(); sx[t] = le + 0.5f * ll + 0.5f * lq; __syncthreads();
    float lt = ((gcfloat)GLb)[t];
    #pragma unroll 8
    for (int j = 0; j < 64; ++j) lt += ((gcfloat)GLw)[t * 64 + j] * sx[j];

    // softmax over C=50 of (kv @ key_matrix.T), stored zero-padded to 64
    auto softmax_out = [&](float kv, float* __restrict__ outp) {
        gcfloat km = (gcfloat)key_matrix;
        __syncthreads();
        kvS[t] = kv;
        __syncthreads();
        float logit = -1e30f;
        if (t < Csz) {
            float z = 0.f;
            #pragma unroll 8
            for (int j = 0; j < 64; ++j) z += kvS[j] * km[t * 64 + j];
            logit = z;
        }
        lg[t] = logit;
        __syncthreads();
        if (t == 0) {
            float mx = -1e30f;
            for (int c = 0; c < Csz; ++c) mx = fmaxf(mx, lg[c]);
            float sm = 0.f;
            for (int c = 0; c < Csz; ++c) sm += expf(lg[c] - mx);
            red[0] = mx; red[1] = sm;
        }
        __syncthreads();
        ((gfloat)outp)[tok * 64 + t] = (t < Csz) ? expf(lg[t] - red[0]) / red[1] : 0.f;
    };

    // w : key from qt (d==0) or lt (d==1)
    __syncthreads(); sx[t] = id ? lt : qt; __syncthreads();
    {
        gcfloat kw = (gcfloat)(id ? mLkw : mQkw);
        gcfloat kb = (gcfloat)(id ? mLkb : mQkb);
        float kv = kb[t];
        #pragma unroll 8
        for (int j = 0; j < 64; ++j) kv += kw[t * 64 + j] * sx[j];
        softmax_out(kv, Wo);
    }
    // w_next : key from raw q embedding
    __syncthreads(); sx[t] = qe; __syncthreads();
    {
        float kv = ((gcfloat)mQkb)[t];
        #pragma unroll 8
        for (int j = 0; j < 64; ++j) kv += ((gcfloat)mQkw)[t * 64 + j] * sx[j];
        softmax_out(kv, WNo);
    }
    // w_type
    __syncthreads(); sx[t] = id ? lt : qt; __syncthreads();
    {
        gcfloat kw = (gcfloat)(id ? mLtw : mQtw);
        gcfloat kb = (gcfloat)(id ? mLtb : mQtb);
        float kv = kb[t];
        #pragma unroll 8
        for (int j = 0; j < 64; ++j) kv += kw[t * 64 + j] * sx[j];
        softmax_out(kv, WTYo);
    }

    // erase / add gates: input is qa=[qt,a_emb] (128) for d==0, lt (64) for d==1
    __syncthreads();
    if (id == 0) { sx[t] = qt; sx[64 + t] = ae; } else { sx[t] = lt; }
    __syncthreads();
    float ev, av;
    if (id == 0) {
        float z1 = ((gcfloat)eQb)[t], z2 = ((gcfloat)aQb)[t];
        #pragma unroll 8
        for (int j = 0; j < 128; ++j) {
            z1 += ((gcfloat)eQw)[t * 128 + j] * sx[j];
            z2 += ((gcfloat)aQw)[t * 128 + j] * sx[j];
        }
        ev = 1.f / (1.f + expf(-z1));
        av = tanhf(z2);
    } else {
        float z1 = ((gcfloat)eLb)[t], z2 = ((gcfloat)aLb)[t];
        #pragma unroll 8
        for (int j = 0; j < 64; ++j) {
            z1 += ((gcfloat)eLw)[t * 64 + j] * sx[j];
            z2 += ((gcfloat)aLw)[t * 64 + j] * sx[j];
        }
        ev = 1.f / (1.f + expf(-z1));
        av = tanhf(z2);
    }
    ((gfloat)Eo) [tok * 64 + t] = ev;
    ((gfloat)ADo)[tok * 64 + t] = av;
    ((gfloat)QEo)[tok * 64 + t] = qe;
}

// -------------------------------------------------------------------------
// Kernel 2: sequential memory scan. One block (4 waves) per batch element.
// V (64x64, rows 50..63 zero) lives in LDS. Per step, wave w computes its
// 16 rows of V @ T_sel.T with V_WMMA_F32_16X16X4_F32 (4 N-tiles x 16 K
// steps, shared A fragment, 4 independent accumulator chains), fuses the
// erase-add write in registers, and stores back in place (each wave only
// touches its own rows). Reads / summary FC / output heads are VALU.
// -------------------------------------------------------------------------
__global__ __launch_bounds__(128) void gmkt_scan(
    const int* __restrict__ ddat,
    const float* __restrict__ TQQ, const float* __restrict__ TQL,
    const float* __restrict__ TLQ, const float* __restrict__ TLL,
    const float* __restrict__ value_init,
    const float* __restrict__ sw,  const float* __restrict__ sb,
    const float* __restrict__ low, const float* __restrict__ lob,
    const float* __restrict__ otqw, const float* __restrict__ otqb,
    const float* __restrict__ otlw, const float* __restrict__ otlb,
    const float* __restrict__ Wbuf, const float* __restrict__ WNbuf,
    const float* __restrict__ WTYbuf, const float* __restrict__ Ebuf,
    const float* __restrict__ ADbuf, const float* __restrict__ QEbuf,
    float* __restrict__ out)
{
    const int b    = blockIdx.x;
    const int tid  = threadIdx.x;
    const int lane = tid & 31;
    const int wv   = tid >> 5;
    const int m    = lane & 15;   // row (A) / col (B) within tile
    const int hi   = lane >> 4;   // lane-half selects K pair / M+8
    const int mbase = wv * 16;

    __shared__ float V[64 * VSTR];
    __shared__ float wtS[64], wnS[64], wtyS[64], eS[64], adS[64];
    __shared__ float xS[128], sS[128], rtS[64];

    for (int i = tid; i < 64 * VSTR; i += 128) V[i] = 0.f;
    __syncthreads();
    for (int i = tid; i < Csz * 64; i += 128) {
        const int c = i >> 6, v = i & 63;
        V[c * VSTR + v] = ((gcfloat)value_init)[i];
    }
    __syncthreads();

    gcfloat Tm[4] = {(gcfloat)TQQ, (gcfloat)TQL, (gcfloat)TLQ, (gcfloat)TLL};
    gcfloat Wg   = (gcfloat)Wbuf,  WNg = (gcfloat)WNbuf, WTYg = (gcfloat)WTYbuf;
    gcfloat Eg   = (gcfloat)Ebuf,  ADg = (gcfloat)ADbuf, QEg  = (gcfloat)QEbuf;
    gcfloat swg  = (gcfloat)sw,    sbg = (gcfloat)sb;
    gcfloat lowg = (gcfloat)low;
    gcint   ddg  = (gcint)ddat;
    gfloat  outg = (gfloat)out;

    for (int t = 0; t < Tlen; ++t) {
        const int s1 = (b * Ssz + t + 1) * 64;
        const int s2 = (b * Ssz + t + 2) * 64;
        if (tid < 64) {
            wtS [tid] = Wg  [s1 + tid];
            wtyS[tid] = WTYg[s1 + tid];
            eS  [tid] = Eg  [s1 + tid];
        } else {
            const int v = tid - 64;
            wnS[v]     = WNg[s2 + v];
            adS[v]     = ADg[s1 + v];
            xS[64 + v] = QEg[s2 + v];   // qn half of summary input
        }
        const int dm = ddg[b * Ssz + t];
        const int dt = ddg[b * Ssz + t + 1];
        gcfloat Tp = Tm[((dm != 0) << 1) | (dt != 0)];
        __syncthreads();

        // ---- matmul phase: rows [mbase, mbase+16) of V1 = V @ Tp^T ----
        v8f acc0 = {}, acc1 = {}, acc2 = {}, acc3 = {};
        #pragma unroll
        for (int k = 0; k < 16; ++k) {
            const int kk = 4 * k + 2 * hi;
            v2f a;                               // A 16x4 fragment (LDS)
            a.x = V[(mbase + m) * VSTR + kk];
            a.y = V[(mbase + m) * VSTR + kk + 1];
            v2f b0, b1, b2, b3;                  // B 4x16 = Tp^T (global)
            b0.x = Tp[( 0 + m) * 64 + kk]; b0.y = Tp[( 0 + m) * 64 + kk + 1];
            b1.x = Tp[(16 + m) * 64 + kk]; b1.y = Tp[(16 + m) * 64 + kk + 1];
            b2.x = Tp[(32 + m) * 64 + kk]; b2.y = Tp[(32 + m) * 64 + kk + 1];
            b3.x = Tp[(48 + m) * 64 + kk]; b3.y = Tp[(48 + m) * 64 + kk + 1];
            acc0 = __builtin_amdgcn_wmma_f32_16x16x4_f32(false, a, false, b0,
                                                         (short)0, acc0, false, false);
            acc1 = __builtin_amdgcn_wmma_f32_16x16x4_f32(false, a, false, b1,
                                                         (short)0, acc1, false, false);
            acc2 = __builtin_amdgcn_wmma_f32_16x16x4_f32(false, a, false, b2,
                                                         (short)0, acc2, false, false);
            acc3 = __builtin_amdgcn_wmma_f32_16x16x4_f32(false, a, false, b3,
                                                         (short)0, acc3, false, false);
        }

        // ---- fused erase-add, write back in place (own rows only) ----
        v8f accs[4] = {acc0, acc1, acc2, acc3};
        #pragma unroll
        for (int nt = 0; nt < 4; ++nt) {
            const int ncol = nt * 16 + m;
            const float ev = eS[ncol];
            const float av = adS[ncol];
            #pragma unroll
            for (int r = 0; r < 8; ++r) {
                const int row = mbase + r + 8 * hi;   // C/D layout: M = r + 8*hi
                const float wr = wtS[row];            // zero for padded rows >=50
                V[row * VSTR + ncol] = accs[nt][r] * (1.f - wr * ev) + wr * av;
            }
        }
        __syncthreads();

        // ---- reads: rn (threads 0..63), rt (threads 64..127) ----
        if (tid < 64) {
            float rn = 0.f;
            for (int c = 0; c < Csz; ++c) rn += wnS[c] * V[c * VSTR + tid];
            xS[tid] = rn;
        } else {
            const int v = tid - 64;
            float rt = 0.f;
            for (int c = 0; c < Csz; ++c) rt += wtyS[c] * V[c * VSTR + v];
            rtS[v] = rt;
        }
        __syncthreads();

        // ---- summary FC (128x128) + tanh ----
        float z = sbg[tid];
        #pragma unroll 8
        for (int j = 0; j < 128; ++j) z += swg[tid * 128 + j] * xS[j];
        sS[tid] = tanhf(z);
        __syncthreads();

        // ---- output heads ----
        if (tid == 0) {
            float p = ((gcfloat)lob)[0];
            for (int j = 0; j < 128; ++j) p += lowg[j] * sS[j];
            outg[(b * Tlen + t) * 2 + 0] = 1.f / (1.f + expf(-p));
        } else if (tid == 1) {
            gcfloat otw = (gcfloat)(dt ? otlw : otqw);
            float p = dt ? ((gcfloat)otlb)[0] : ((gcfloat)otqb)[0];
            for (int v = 0; v < 64; ++v) p += otw[v] * rtS[v];
            outg[(b * Tlen + t) * 2 + 1] = 1.f / (1.f + expf(-p));
        }
        __syncthreads();
    }
}

// -------------------------------------------------------------------------
extern "C" void kernel_launch(void* const* d_in, const int* in_sizes, int n_in,
                              void* d_out, int out_size, void* d_ws, size_t ws_size,
                              hipStream_t stream) {
    (void)in_sizes; (void)n_in; (void)out_size; (void)ws_size;
    const int*   qd     = (const int*)  d_in[0];
    const int*   adat   = (const int*)  d_in[1];
    const int*   ldat   = (const int*)  d_in[2];
    const int*   ddat   = (const int*)  d_in[3];
    const int*   qqn    = (const int*)  d_in[4];
    const int*   qln    = (const int*)  d_in[5];
    const int*   lqn    = (const int*)  d_in[6];
    const int*   lln    = (const int*)  d_in[7];
    const float* q_emb  = (const float*)d_in[8];
    const float* l_emb  = (const float*)d_in[9];
    const float* a_emb  = (const float*)d_in[10];
    const float* keym   = (const float*)d_in[11];
    const float* vinit  = (const float*)d_in[12];
    const float* GQQ    = (const float*)d_in[13];
    const float* GQL    = (const float*)d_in[14];
    const float* GLL    = (const float*)d_in[15];
    const float* GLQ    = (const float*)d_in[16];
    const float* TQQ    = (const float*)d_in[17];
    const float* TQL    = (const float*)d_in[18];
    const float* TLQ    = (const float*)d_in[19];
    const float* TLL    = (const float*)d_in[20];
    const float* GQw    = (const float*)d_in[21];
    const float* GQb    = (const float*)d_in[22];
    const float* GLw    = (const float*)d_in[23];
    const float* GLb    = (const float*)d_in[24];
    const float* mQkw   = (const float*)d_in[25];
    const float* mQkb   = (const float*)d_in[26];
    const float* mLkw   = (const float*)d_in[27];
    const float* mLkb   = (const float*)d_in[28];
    const float* mQtw   = (const float*)d_in[29];
    const float* mQtb   = (const float*)d_in[30];
    const float* mLtw   = (const float*)d_in[31];
    const float* mLtb   = (const float*)d_in[32];
    const float* eQw    = (const float*)d_in[33];
    const float* eQb    = (const float*)d_in[34];
    const float* eLw    = (const float*)d_in[35];
    const float* eLb    = (const float*)d_in[36];
    const float* aQw    = (const float*)d_in[37];
    const float* aQb    = (const float*)d_in[38];
    const float* aLw    = (const float*)d_in[39];
    const float* aLb    = (const float*)d_in[40];
    const float* sw     = (const float*)d_in[41];
    const float* sb     = (const float*)d_in[42];
    const float* low    = (const float*)d_in[43];
    const float* lob    = (const float*)d_in[44];
    const float* otqw   = (const float*)d_in[45];
    const float* otqb   = (const float*)d_in[46];
    const float* otlw   = (const float*)d_in[47];
    const float* otlb   = (const float*)d_in[48];

    float* ws = (float*)d_ws;
    const size_t TOK = (size_t)Bsz * Ssz;
    float* Wo   = ws;
    float* WNo  = Wo   + TOK * 64;
    float* WTYo = WNo  + TOK * 64;
    float* Eo   = WTYo + TOK * 64;
    float* ADo  = Eo   + TOK * 64;
    float* QEo  = ADo  + TOK * 64;   // total 6 * 12800 * 64 floats ~ 19.7 MB

    gmkt_pre<<<dim3((unsigned)TOK), dim3(64), 0, stream>>>(
        qd, adat, ldat, ddat, qqn, qln, lqn, lln,
        q_emb, l_emb, a_emb, keym,
        GQQ, GQL, GLL, GLQ,
        GQw, GQb, GLw, GLb,
        mQkw, mQkb, mLkw, mLkb,
        mQtw, mQtb, mLtw, mLtb,
        eQw, eQb, eLw, eLb,
        aQw, aQb, aLw, aLb,
        Wo, WNo, WTYo, Eo, ADo, QEo);

    gmkt_scan<<<dim3(Bsz), dim3(128), 0, stream>>>(
        ddat, TQQ, TQL, TLQ, TLL, vinit,
        sw, sb, low, lob, otqw, otqb, otlw, otlb,
        Wo, WNo, WTYo, Eo, ADo, QEo, (float*)d_out);
}